// SelectiveSSM_76656576299294
// MI455X (gfx1250) — compile-verified
//
#include <hip/hip_runtime.h>
#include <math.h>

// ---------------------------------------------------------------------------
// Selective SSM (Mamba block) forward for MI455X / gfx1250 (wave32, WMMA).
//
// Shapes: B=2, L=1024, d_model=1024, d_inner=2048, d_state=16, d_conv=4.
// "Rows" below means flattened B*L = 2048 rows.
//
// Pipeline:
//   1) xz     = x @ W_in                       [2048 x 4096]   (WMMA f32)
//   2) x_conv = silu(causal_dwconv(xz[:, :2048]))  [2048 x 2048]
//   3) x_dbl  = x_conv @ W_xproj               [2048 x 2080]   (WMMA f32)
//   4) delta  = softplus(x_dbl[:, :2048] @ W_dt + b_dt)        (WMMA f32)
//   5) scan   : state recurrence over t, thread-per-(chan,state),
//               16-lane shuffle reduction for the C-contraction
//   6) y     *= silu(z)          (z = xz[:, 2048:])
//   7) out    = y @ W_out                      [2048 x 1024]   (WMMA f32)
// ---------------------------------------------------------------------------

#define D_MODEL 1024
#define D_STATE 16
#define D_CONV  4
#define D_INNER 2048
#define BATCH   2
#define SEQ     1024
#define ROWS    (BATCH * SEQ)           // 2048
#define N_XPROJ (D_INNER + 2 * D_STATE) // 2080

typedef float v2f __attribute__((ext_vector_type(2)));
typedef float v8f __attribute__((ext_vector_type(8)));

#define WMMA_F32(a, b, c)                                               \
    __builtin_amdgcn_wmma_f32_16x16x4_f32(false, (a), false, (b),       \
                                          (short)0, (c), false, false)

// ---------------------------------------------------------------------------
// fp32 WMMA GEMM: C[M,N] = A[M,K] * B[K,N]  (+bias, +softplus for act==1)
// Block: 128 threads = 4 waves stacked in M; each wave owns a 16x64 strip
// (4 accumulators), so each A fragment is reused by 4 WMMAs.
// Block tile = 64 x 64.  Grid: (M/64, ceil(N/64)); N always a multiple of 16.
// Per the CDNA5 ISA 16x16x4 f32 layouts:
//   A frag (2 VGPR):  lane l -> row m = l&15,  k_local = (l>>4)*2 + v
//   B frag (2 VGPR):  lane l -> col n = l&15,  k_local = (l>>4)*2 + v
//   C/D   (8 VGPR):   lane l, vgpr v -> row = v + (l>>4)*8, col = l&15
// Tail N-subtiles clamp load columns branchlessly (EXEC stays all-ones for
// WMMA) and guard stores with a wave-uniform test.
// ---------------------------------------------------------------------------
__global__ __launch_bounds__(128) void ssm_gemm_wmma_f32(
    const float* __restrict__ A, const float* __restrict__ B,
    float* __restrict__ C, int M, int N, int K,
    int lda, int ldb, int ldc,
    const float* __restrict__ bias, int act)
{
    const int wave = threadIdx.x >> 5;      // 0..3
    const int lane = threadIdx.x & 31;
    const int m0   = blockIdx.x * 64 + wave * 16;
    const int n0   = blockIdx.y * 64;

    const int lo16  = lane & 15;            // row in A frag / col in B frag
    const int khalf = (lane >> 4) * 2;      // 0 (lanes 0-15) or 2 (lanes 16-31)

    // A: row gather, khalf folded into the pointer
    const float* __restrict__ Aptr = A + (long)(m0 + lo16) * lda + khalf;

    // Per-subtile B column bases; clamp out-of-range columns to stay
    // in-bounds (results discarded at store time).
    int c0 = n0 + lo16;       if (c0 >= N) c0 = N - 1;
    int c1 = n0 + 16 + lo16;  if (c1 >= N) c1 = N - 1;
    int c2 = n0 + 32 + lo16;  if (c2 >= N) c2 = N - 1;
    int c3 = n0 + 48 + lo16;  if (c3 >= N) c3 = N - 1;
    const float* __restrict__ B0 = B + c0;
    const float* __restrict__ B1 = B + c1;
    const float* __restrict__ B2 = B + c2;
    const float* __restrict__ B3 = B + c3;

    v8f acc0 = {}, acc1 = {}, acc2 = {}, acc3 = {};

    #pragma unroll 2
    for (int k = 0; k < K; k += 4) {
        v2f a;
        a.x = Aptr[k];
        a.y = Aptr[k + 1];

        const long off  = (long)(k + khalf) * ldb;
        const long off1 = off + ldb;
        v2f b0, b1, b2, b3;
        b0.x = B0[off]; b0.y = B0[off1];
        b1.x = B1[off]; b1.y = B1[off1];
        b2.x = B2[off]; b2.y = B2[off1];
        b3.x = B3[off]; b3.y = B3[off1];

        acc0 = WMMA_F32(a, b0, acc0);
        acc1 = WMMA_F32(a, b1, acc1);
        acc2 = WMMA_F32(a, b2, acc2);
        acc3 = WMMA_F32(a, b3, acc3);
    }

    const int mbase = m0 + (lane >> 4) * 8;

#define SSM_STORE_TILE(j, accv)                                              \
    if (n0 + 16 * (j) < N) { /* wave-uniform; N % 16 == 0 */                 \
        const int cn = n0 + 16 * (j) + lo16;                                 \
        const float bn = (act == 1) ? bias[cn] : 0.0f;                       \
        _Pragma("unroll")                                                    \
        for (int v = 0; v < 8; ++v) {                                        \
            float val = accv[v];                                             \
            if (act == 1) {                                                  \
                val += bn;                                /* + b_dt */       \
                val = (val > 20.0f) ? val : log1pf(__expf(val)); /*softplus*/\
            }                                                                \
            C[(long)(mbase + v) * ldc + cn] = val;                           \
        }                                                                    \
    }

    SSM_STORE_TILE(0, acc0)
    SSM_STORE_TILE(1, acc1)
    SSM_STORE_TILE(2, acc2)
    SSM_STORE_TILE(3, acc3)
#undef SSM_STORE_TILE
}

// ---------------------------------------------------------------------------
// Causal depthwise conv1d (kernel 4, left zero-pad 3) + bias + SiLU.
// Input is the x-half of xz (row stride 4096); output [ROWS x D_INNER].
// ---------------------------------------------------------------------------
__global__ __launch_bounds__(256) void ssm_conv_silu(
    const float* __restrict__ xz, const float* __restrict__ conv_w,
    const float* __restrict__ conv_b, float* __restrict__ xconv)
{
    int idx = blockIdx.x * blockDim.x + threadIdx.x;   // over ROWS*D_INNER
    if (idx >= ROWS * D_INNER) return;
    int c   = idx & (D_INNER - 1);
    int row = idx >> 11;                               // b*SEQ + t
    int t   = row & (SEQ - 1);

    float acc = conv_b[c];
    const float* w = conv_w + c * D_CONV;
    #pragma unroll
    for (int k = 0; k < D_CONV; ++k) {
        int tt = t + k - (D_CONV - 1);
        if (tt >= 0)
            acc = fmaf(xz[(long)(row + k - (D_CONV - 1)) * (2 * D_INNER) + c],
                       w[k], acc);
    }
    acc = acc / (1.0f + __expf(-acc));                 // silu
    xconv[idx] = acc;
}

// ---------------------------------------------------------------------------
// Selective scan. One thread per (batch, channel, state): 65536 threads =
// 2048 wave32s. 16 consecutive lanes share a channel; reduce sum(state*C)
// over the 16 states with a shuffle butterfly. 1024 sequential steps per
// thread; pointer-bumped streams + global_prefetch to hide the stride-8KB
// access latency under the serial dependence chain.
// ---------------------------------------------------------------------------
__global__ __launch_bounds__(256) void ssm_scan(
    const float* __restrict__ xconv, const float* __restrict__ delta,
    const float* __restrict__ xdbl,  const float* __restrict__ A_log,
    const float* __restrict__ Dp,    float* __restrict__ y)
{
    int tid = threadIdx.x;
    int s   = tid & (D_STATE - 1);                 // state index 0..15
    int cg  = blockIdx.x * 16 + (tid >> 4);        // global channel 0..4095
    int b   = cg >> 11;                            // batch
    int c   = cg & (D_INNER - 1);                  // channel in d_inner

    const float Acs = -__expf(A_log[c * D_STATE + s]);   // A = -exp(A_log)
    const float Dc  = Dp[c];

    const long rb = (long)b * SEQ;
    const float* __restrict__ xr = xconv + rb * D_INNER + c;
    const float* __restrict__ dr = delta + rb * D_INNER + c;
    const float* __restrict__ br = xdbl  + rb * N_XPROJ + D_INNER + s;
    const float* __restrict__ cr = xdbl  + rb * N_XPROJ + D_INNER + D_STATE + s;
    float* __restrict__       yr = y     + rb * D_INNER + c;

    float state = 0.0f;
    for (int t = 0; t < SEQ; ++t) {
        // pull rows 4 iterations ahead toward the WGP while we compute
        __builtin_prefetch(xr + 4 * D_INNER, 0, 0);
        __builtin_prefetch(dr + 4 * D_INNER, 0, 0);

        float xt = *xr;
        float dt = *dr;
        float Bt = *br;
        float Ct = *cr;

        float decay = __expf(dt * Acs);
        state = fmaf(decay, state, Bt * xt);

        float ps = state * Ct;
        ps += __shfl_xor(ps, 1);
        ps += __shfl_xor(ps, 2);
        ps += __shfl_xor(ps, 4);
        ps += __shfl_xor(ps, 8);
        if (s == 0)
            *yr = ps + xt * Dc;

        xr += D_INNER; dr += D_INNER; yr += D_INNER;
        br += N_XPROJ; cr += N_XPROJ;
    }
}

// ---------------------------------------------------------------------------
// y *= silu(z), z = xz[:, 2048:]
// ---------------------------------------------------------------------------
__global__ __launch_bounds__(256) void ssm_gate(
    const float* __restrict__ xz, float* __restrict__ y)
{
    int idx = blockIdx.x * blockDim.x + threadIdx.x;
    if (idx >= ROWS * D_INNER) return;
    int c   = idx & (D_INNER - 1);
    int row = idx >> 11;
    float z  = xz[(long)row * (2 * D_INNER) + D_INNER + c];
    float sz = z / (1.0f + __expf(-z));
    y[idx] *= sz;
}

// ---------------------------------------------------------------------------
extern "C" void kernel_launch(void* const* d_in, const int* in_sizes, int n_in,
                              void* d_out, int out_size, void* d_ws, size_t ws_size,
                              hipStream_t stream)
{
    const float* x      = (const float*)d_in[0];   // [2,1024,1024]
    const float* W_in   = (const float*)d_in[1];   // [1024,4096]
    const float* conv_w = (const float*)d_in[2];   // [2048,4]
    const float* conv_b = (const float*)d_in[3];   // [2048]
    const float* W_xprj = (const float*)d_in[4];   // [2048,2080]
    const float* W_dt   = (const float*)d_in[5];   // [2048,2048]
    const float* b_dt   = (const float*)d_in[6];   // [2048]
    const float* A_log  = (const float*)d_in[7];   // [2048,16]
    const float* Dp     = (const float*)d_in[8];   // [2048]
    const float* W_out  = (const float*)d_in[9];   // [2048,1024]
    float*       out    = (float*)d_out;           // [2,1024,1024]

    // workspace carve-up (floats), every buffer fully written before read
    float* xz    = (float*)d_ws;                         // 2048*4096
    float* xconv = xz    + (long)ROWS * 2 * D_INNER;     // 2048*2048
    float* xdbl  = xconv + (long)ROWS * D_INNER;         // 2048*2080
    float* delta = xdbl  + (long)ROWS * N_XPROJ;         // 2048*2048
    float* y     = delta + (long)ROWS * D_INNER;         // 2048*2048

    dim3 blk128(128), blk256(256);

    // 1) xz = x @ W_in              M=2048 N=4096 K=1024
    ssm_gemm_wmma_f32<<<dim3(ROWS / 64, (2 * D_INNER) / 64), blk128, 0, stream>>>(
        x, W_in, xz, ROWS, 2 * D_INNER, D_MODEL,
        D_MODEL, 2 * D_INNER, 2 * D_INNER, nullptr, 0);

    // 2) x_conv = silu(conv(x_inner))
    ssm_conv_silu<<<dim3((ROWS * D_INNER) / 256), blk256, 0, stream>>>(
        xz, conv_w, conv_b, xconv);

    // 3) x_dbl = x_conv @ W_xproj   M=2048 N=2080 K=2048  (N tail: 2080=32*65)
    ssm_gemm_wmma_f32<<<dim3(ROWS / 64, (N_XPROJ + 63) / 64), blk128, 0, stream>>>(
        xconv, W_xprj, xdbl, ROWS, N_XPROJ, D_INNER,
        D_INNER, N_XPROJ, N_XPROJ, nullptr, 0);

    // 4) delta = softplus(x_dbl[:, :2048] @ W_dt + b_dt)   (A has lda=2080)
    ssm_gemm_wmma_f32<<<dim3(ROWS / 64, D_INNER / 64), blk128, 0, stream>>>(
        xdbl, W_dt, delta, ROWS, D_INNER, D_INNER,
        N_XPROJ, D_INNER, D_INNER, b_dt, 1);

    // 5) selective scan -> y
    ssm_scan<<<dim3((BATCH * D_INNER) / 16), blk256, 0, stream>>>(
        xconv, delta, xdbl, A_log, Dp, y);

    // 6) y *= silu(z)
    ssm_gate<<<dim3((ROWS * D_INNER) / 256), blk256, 0, stream>>>(xz, y);

    // 7) out = y @ W_out            M=2048 N=1024 K=2048
    ssm_gemm_wmma_f32<<<dim3(ROWS / 64, D_MODEL / 64), blk128, 0, stream>>>(
        y, W_out, out, ROWS, D_MODEL, D_INNER,
        D_INNER, D_MODEL, D_MODEL, nullptr, 0);
}